// NCGM_13400297964072
// MI455X (gfx1250) — compile-verified
//
#include <hip/hip_runtime.h>

typedef __attribute__((ext_vector_type(2))) float v2f;
typedef __attribute__((ext_vector_type(4))) float v4f;
typedef __attribute__((ext_vector_type(8))) float v8f;

#define NB     2
#define NN     160
#define HH     256
#define C3     768          // 3*H
#define C6     1536         // 2*C3
#define FF     512          // hidden
#define MROWS  (NB * NN)    // 320 rows of emb
#define SCOLS  (6 * FF)     // 3072: [cell_top, cell_bot, row_top, row_bot, col_top, col_bot]
#define PAIRS  (NN * NN)    // 25600 pairs per batch

// ---------------------------------------------------------------------------
// Kernel 0: materialize emb = concat(geometry, appearance, content) (320 x 768)
// ---------------------------------------------------------------------------
__global__ void k_gather(const float* __restrict__ g, const float* __restrict__ a,
                         const float* __restrict__ c, float* __restrict__ emb) {
    int idx = blockIdx.x * blockDim.x + threadIdx.x;
    if (idx >= MROWS * C3) return;
    int m = idx / C3;
    int f = idx - m * C3;
    const float* src = (f < HH) ? g : (f < 2 * HH ? a : c);
    int off = (f < HH) ? f : (f < 2 * HH ? f - HH : f - 2 * HH);
    emb[idx] = src[m * HH + off];
}

// ---------------------------------------------------------------------------
// Kernel 1: S = emb @ [W1 halves], fp32 WMMA 16x16x4. One wave per 16x16 tile.
//   A 16x4 layout : lanes 0-15 = M, vgpr0 = {K=0 | K=2}, vgpr1 = {K=1 | K=3}
//   B 4x16 layout : lanes = N,    vgpr0 = {K=0 | K=2}, vgpr1 = {K=1 | K=3}
//   D 16x16 layout: vgpr r = row (r | r+8), col = lane & 15
// ---------------------------------------------------------------------------
__global__ void k_gemm1(const float* __restrict__ emb,
                        const float* __restrict__ w_cell,
                        const float* __restrict__ w_row,
                        const float* __restrict__ w_col,
                        float* __restrict__ S) {
    const int MT = MROWS / 16;            // 20 row tiles
    const int CT = SCOLS / 16;            // 192 col tiles
    int wave = (blockIdx.x * blockDim.x + threadIdx.x) >> 5;
    if (wave >= MT * CT) return;
    int lane = threadIdx.x & 31;
    int half = lane >> 4;                 // 0: lanes 0-15, 1: lanes 16-31
    int l16  = lane & 15;

    int mt = wave % MT;
    int ct = wave / MT;
    int m0   = mt * 16;
    int scol = ct * 16;
    int grp  = scol / FF;                 // 0..5
    int head = grp >> 1;
    int bot  = grp & 1;
    const float* W1 = (head == 0) ? w_cell : (head == 1 ? w_row : w_col);
    const float* Wb = W1 + (size_t)bot * C3 * FF;   // (768 x 512) block, row-major
    int ncol = (scol % FF) + l16;

    const float* arow = emb + (size_t)(m0 + l16) * C3;
    v8f acc = {};
    for (int k = 0; k < C3; k += 4) {
        int ka = k + 2 * half;
        v2f av; av.x = arow[ka];                 // adjacent -> b64 load
                av.y = arow[ka + 1];
        v2f bv; bv.x = Wb[(size_t)ka * FF + ncol];        // coalesced across lanes
                bv.y = Wb[(size_t)(ka + 1) * FF + ncol];
        acc = __builtin_amdgcn_wmma_f32_16x16x4_f32(
                  false, av, false, bv, (short)0, acc, false, false);
    }
    #pragma unroll
    for (int r = 0; r < 8; ++r) {
        int row = m0 + r + 8 * half;
        S[(size_t)row * SCOLS + scol + l16] = acc[r];
    }
}

// ---------------------------------------------------------------------------
// Kernel 2: per 16x16 pair tile, 3 heads:
//   z_k = b2_k + sum_f relu(Stop[j,f] + Sbot[i,f] + b1[f]) * W2[f,k]; softmax.
// ---------------------------------------------------------------------------
#define FCH 128
__global__ void k_heads(const float* __restrict__ S,
                        const float* __restrict__ b1_0, const float* __restrict__ W2_0, const float* __restrict__ b2_0,
                        const float* __restrict__ b1_1, const float* __restrict__ W2_1, const float* __restrict__ b2_1,
                        const float* __restrict__ b1_2, const float* __restrict__ W2_2, const float* __restrict__ b2_2,
                        float* __restrict__ out) {
    __shared__ float stop[16][FCH + 1];   // +1 pad: bank-conflict-free column reads
    __shared__ float sbot[16][FCH + 1];
    __shared__ float b1c[FCH];
    __shared__ float w2c[FCH][2];

    int b  = blockIdx.z;
    int i0 = blockIdx.y * 16;
    int j0 = blockIdx.x * 16;
    int tid = threadIdx.x;
    int tj = tid & 15, ti = tid >> 4;

    const float* b1s[3] = { b1_0, b1_1, b1_2 };
    const float* W2s[3] = { W2_0, W2_1, W2_2 };
    const float* b2s[3] = { b2_0, b2_1, b2_2 };

    for (int t = 0; t < 3; ++t) {
        float z0 = b2s[t][0], z1 = b2s[t][1];
        for (int f0 = 0; f0 < FF; f0 += FCH) {
            __syncthreads();
            for (int u = tid; u < 16 * FCH; u += 256) {
                int r = u / FCH, c = u - r * FCH;
                stop[r][c] = S[(size_t)(b * NN + j0 + r) * SCOLS + (2 * t)     * FF + f0 + c];
                sbot[r][c] = S[(size_t)(b * NN + i0 + r) * SCOLS + (2 * t + 1) * FF + f0 + c];
            }
            if (tid < FCH) {
                b1c[tid]    = b1s[t][f0 + tid];
                w2c[tid][0] = W2s[t][(f0 + tid) * 2 + 0];
                w2c[tid][1] = W2s[t][(f0 + tid) * 2 + 1];
            }
            __syncthreads();
            #pragma unroll 8
            for (int c = 0; c < FCH; ++c) {
                float h = fmaxf(stop[tj][c] + sbot[ti][c] + b1c[c], 0.0f);
                z0 = fmaf(h, w2c[c][0], z0);
                z1 = fmaf(h, w2c[c][1], z1);
            }
        }
        float mx = fmaxf(z0, z1);
        float e0 = __expf(z0 - mx), e1 = __expf(z1 - mx);
        float inv = 1.0f / (e0 + e1);
        size_t p = (size_t)b * PAIRS + (size_t)(i0 + ti) * NN + (j0 + tj);
        float* o = out + (size_t)t * ((size_t)NB * PAIRS * 2) + p * 2;
        o[0] = e0 * inv;
        o[1] = e1 * inv;
    }
}

// ---------------------------------------------------------------------------
// Kernel 3: emb_pairs[b, i*N+j] = [emb[b,j], emb[b,i]]  (314 MB, NT stores)
// ---------------------------------------------------------------------------
__global__ void k_pairs(const float* __restrict__ emb, float* __restrict__ outp) {
    const size_t total = (size_t)NB * PAIRS * (C6 / 4);   // float4 count
    size_t idx = (size_t)blockIdx.x * blockDim.x + threadIdx.x;
    if (idx >= total) return;
    int    q    = (int)(idx % (C6 / 4));                  // 0..383
    size_t pair = idx / (C6 / 4);
    int j = (int)(pair % NN);
    int i = (int)((pair / NN) % NN);
    int b = (int)(pair / PAIRS);
    int f = q * 4;
    int srcrow = (f < C3) ? (b * NN + j) : (b * NN + i);
    int fo     = (f < C3) ? f : (f - C3);
    v4f val = *(const v4f*)(emb + (size_t)srcrow * C3 + fo);
    __builtin_nontemporal_store(val, (v4f*)outp + idx);
}

// ---------------------------------------------------------------------------
extern "C" void kernel_launch(void* const* d_in, const int* in_sizes, int n_in,
                              void* d_out, int out_size, void* d_ws, size_t ws_size,
                              hipStream_t stream) {
    (void)in_sizes; (void)n_in; (void)out_size; (void)ws_size;
    const float* geometry   = (const float*)d_in[0];
    const float* appearance = (const float*)d_in[1];
    const float* content    = (const float*)d_in[2];
    const float* cell_W1 = (const float*)d_in[3];
    const float* cell_b1 = (const float*)d_in[4];
    const float* cell_W2 = (const float*)d_in[5];
    const float* cell_b2 = (const float*)d_in[6];
    const float* row_W1  = (const float*)d_in[7];
    const float* row_b1  = (const float*)d_in[8];
    const float* row_W2  = (const float*)d_in[9];
    const float* row_b2  = (const float*)d_in[10];
    const float* col_W1  = (const float*)d_in[11];
    const float* col_b1  = (const float*)d_in[12];
    const float* col_W2  = (const float*)d_in[13];
    const float* col_b2  = (const float*)d_in[14];

    float* out = (float*)d_out;
    float* emb_ws = (float*)d_ws;                       // 320*768 floats
    float* S_ws   = emb_ws + (size_t)MROWS * C3;        // 320*3072 floats

    // 0) emb gather
    {
        int n = MROWS * C3;
        k_gather<<<(n + 255) / 256, 256, 0, stream>>>(geometry, appearance, content, emb_ws);
    }
    // 1) fp32 WMMA projections S = emb @ {W1 halves}
    {
        int tiles  = (MROWS / 16) * (SCOLS / 16);       // 3840 waves
        int blocks = (tiles + 7) / 8;                   // 8 waves / 256-thread block
        k_gemm1<<<blocks, 256, 0, stream>>>(emb_ws, cell_W1, row_W1, col_W1, S_ws);
    }
    // 2) fused pair MLP + softmax -> three logits blocks
    {
        dim3 grid(NN / 16, NN / 16, NB);                // (10,10,2)
        k_heads<<<grid, 256, 0, stream>>>(S_ws,
                                          cell_b1, cell_W2, cell_b2,
                                          row_b1,  row_W2,  row_b2,
                                          col_b1,  col_W2,  col_b2,
                                          out);
    }
    // 3) emb_pairs stream (after the 3 logits blocks in d_out)
    {
        float* outp = out + (size_t)3 * NB * PAIRS * 2; // offset 307200
        size_t total = (size_t)NB * PAIRS * (C6 / 4);   // 19,660,800 float4s
        int blocks = (int)((total + 255) / 256);
        k_pairs<<<blocks, 256, 0, stream>>>(emb_ws, outp);
    }
}